// SelectiveScan_85237920957164
// MI455X (gfx1250) — compile-verified
//
#include <hip/hip_runtime.h>

#define BATCH 2
#define CH 2048
#define ST 16
#define LEN 2048
#define TCH 64                 // timesteps per LDS chunk
#define NCHUNK (LEN / TCH)     // 32
#define ROWPAD 68              // 64 + 4: 16B-aligned float4 slots, conflict-free columns

typedef __attribute__((ext_vector_type(2))) float v2f;
typedef __attribute__((ext_vector_type(4))) float v4f;
typedef __attribute__((ext_vector_type(8))) float v8f;
typedef __attribute__((ext_vector_type(4))) int   v4i;

#if defined(__HIP_DEVICE_COMPILE__) && defined(__gfx1250__) && \
    __has_builtin(__builtin_amdgcn_global_load_async_to_lds_b128) && \
    __has_builtin(__builtin_amdgcn_s_wait_asynccnt)
#define USE_ASYNC 1
#else
#define USE_ASYNC 0
#endif

__device__ __forceinline__ float fast_rcp(float x) {
#if defined(__HIP_DEVICE_COMPILE__) && __has_builtin(__builtin_amdgcn_rcpf)
  return __builtin_amdgcn_rcpf(x);     // single v_rcp_f32
#else
  return 1.f / x;
#endif
}

__device__ __forceinline__ void cp16(const float* g, float* l) {
#if USE_ASYNC
  __builtin_amdgcn_global_load_async_to_lds_b128(
      (__attribute__((address_space(1))) v4i*)g,
      (__attribute__((address_space(3))) v4i*)l,
      0, 0);
#else
  *(v4f*)l = *(const v4f*)g;
#endif
}

__device__ __forceinline__ void wait_stage() {
#if USE_ASYNC
  __builtin_amdgcn_s_wait_asynccnt(0);
  __asm__ volatile("" ::: "memory");
#endif
}

// Copy a 16-row x 64-col f32 tile (row stride LEN in global) into LDS [16][ROWPAD].
__device__ __forceinline__ void stage_tile(const float* __restrict__ g,
                                           float (*dst)[ROWPAD], int lane) {
#pragma unroll
  for (int i = 0; i < 8; ++i) {
    int idx = (i << 5) + lane;          // 0..255
    int row = idx >> 4;                 // 0..15
    int col = (idx & 15) << 2;          // 0..60 step 4
    cp16(g + (size_t)row * LEN + col, &dst[row][col]);
  }
}

__global__ __launch_bounds__(32)
void SelectiveScan_kernel(const float* __restrict__ u, const float* __restrict__ delta,
                          const float* __restrict__ A, const float* __restrict__ Bm,
                          const float* __restrict__ Cm, const float* __restrict__ Dv,
                          const float* __restrict__ z, const float* __restrict__ dbias,
                          float* __restrict__ out) {
  __shared__ float su[2][16][ROWPAD];
  __shared__ float sd[2][16][ROWPAD];
  __shared__ float sz[2][16][ROWPAD];
  __shared__ float sB[2][16][ROWPAD];
  __shared__ float sC[2][16][ROWPAD];

  const int lane = threadIdx.x;
  const int half = lane >> 4;          // 0/1
  const int dl   = lane & 15;          // channel-in-group = WMMA M / N position
  const int batch = blockIdx.x >> 7;   // 256 blocks = 2 batches x 128 channel groups
  const int d0    = (blockIdx.x & 127) << 4;

  const size_t chanBase = ((size_t)batch * CH + d0) * (size_t)LEN;
  const size_t stBase   = ((size_t)batch * ST) * (size_t)LEN;
  const float* up = u + chanBase;
  const float* dp = delta + chanBase;
  const float* zp = z + chanBase;
  const float* bp = Bm + stBase;
  const float* cp = Cm + stBase;
  float* yp = out + chanBase;

  const float bias = dbias[d0 + dl];
  float Areg[8], Dp[8];
#pragma unroll
  for (int r = 0; r < 8; ++r) {
    // state index held in x[r] for this lane (WMMA A-matrix 16x4 layout, 4 K-slices)
    int n = ((r >> 1) << 2) + (half << 1) + (r & 1);
    Areg[r] = A[(size_t)(d0 + dl) * ST + n];
    // channel handled by accumulator row r in C/D layout
    Dp[r] = Dv[d0 + r + (half << 3)];
  }

  float x[8];
#pragma unroll
  for (int r = 0; r < 8; ++r) x[r] = 0.f;

  // stage chunk 0
  stage_tile(up, su[0], lane);
  stage_tile(dp, sd[0], lane);
  stage_tile(zp, sz[0], lane);
  stage_tile(bp, sB[0], lane);
  stage_tile(cp, sC[0], lane);
  wait_stage();

#pragma unroll 1
  for (int c = 0; c < NCHUNK; ++c) {
    const int buf = c & 1;
    if (c + 1 < NCHUNK) {              // prefetch next chunk into other buffer
      const int t1 = (c + 1) * TCH;
      const int nb = buf ^ 1;
      stage_tile(up + t1, su[nb], lane);
      stage_tile(dp + t1, sd[nb], lane);
      stage_tile(zp + t1, sz[nb], lane);
      stage_tile(bp + t1, sB[nb], lane);
      stage_tile(cp + t1, sC[nb], lane);
    }
    const int t0 = c * TCH;

#pragma unroll 1
    for (int w = 0; w < TCH / 16; ++w) {
      v8f acc[4];
#pragma unroll
      for (int i = 0; i < 4; ++i) acc[i] = (v8f){0.f,0.f,0.f,0.f,0.f,0.f,0.f,0.f};

#pragma unroll
      for (int j = 0; j < 16; ++j) {
        const int tc = (w << 4) + j;
        float del = sd[buf][dl][tc];
        float uu  = su[buf][dl][tc];
        float s   = del + bias;
        float dt  = (s > 15.f) ? s : __logf(1.f + __expf(s));   // softplus
        float dtu = dt * uu;
        const bool hit = (dl == j);     // C_t lands in output column N = j

#pragma unroll
        for (int i = 0; i < 4; ++i) {   // 4 K-slices of the 16-wide state
          int n0 = (i << 2) + (half << 1);
          float b0 = sB[buf][n0][tc];
          float b1 = sB[buf][n0 + 1][tc];
          float c0 = sC[buf][n0][tc];
          float c1 = sC[buf][n0 + 1][tc];
          float xa = __expf(dt * Areg[2 * i])     * x[2 * i]     + dtu * b0;
          float xb = __expf(dt * Areg[2 * i + 1]) * x[2 * i + 1] + dtu * b1;
          x[2 * i] = xa;
          x[2 * i + 1] = xb;
          v2f av; av.x = xa; av.y = xb;                 // A-matrix 16x4 slice
          v2f bv; bv.x = hit ? c0 : 0.f;                // B-matrix 4x16, col j only
          bv.y = hit ? c1 : 0.f;
          acc[i] = __builtin_amdgcn_wmma_f32_16x16x4_f32(
              false, av, false, bv, (short)0, acc[i], false, false);
        }
      }

      v8f yacc = acc[0] + acc[1] + acc[2] + acc[3];     // 16(ch) x 16(t) tile

      // epilogue: y = (y + D*u) * silu(z); lane holds column t = t0 + w*16 + dl
      const int tl = (w << 4) + dl;
#pragma unroll
      for (int r = 0; r < 8; ++r) {
        int row = r + (half << 3);
        float uu = su[buf][row][tl];
        float zz = sz[buf][row][tl];
        float val = yacc[r] + Dp[r] * uu;
        float sig = fast_rcp(1.f + __expf(-zz));        // v_rcp_f32, not IEEE div
        val *= zz * sig;
        yp[(size_t)row * LEN + (size_t)(t0 + tl)] = val;
      }
    }

    if (c + 1 < NCHUNK) wait_stage();
  }
}

extern "C" void kernel_launch(void* const* d_in, const int* in_sizes, int n_in,
                              void* d_out, int out_size, void* d_ws, size_t ws_size,
                              hipStream_t stream) {
  const float* u     = (const float*)d_in[0];
  const float* delta = (const float*)d_in[1];
  const float* A     = (const float*)d_in[2];
  const float* B     = (const float*)d_in[3];
  const float* C     = (const float*)d_in[4];
  const float* D     = (const float*)d_in[5];
  const float* z     = (const float*)d_in[6];
  const float* db    = (const float*)d_in[7];
  float* y = (float*)d_out;

  dim3 grid(BATCH * (CH / 16));   // 256 workgroups, one wave32 each
  dim3 block(32);
  hipLaunchKernelGGL(SelectiveScan_kernel, grid, block, 0, stream,
                     u, delta, A, B, C, D, z, db, y);
}